// SpatialPyramidPool_1958505087113
// MI455X (gfx1250) — compile-verified
//
#include <hip/hip_runtime.h>

// SpatialPyramidPool, channels_first, levels {1,2,4,8}, average pooling over
// (C, bin_h, bin_w). x: [16, 64, 256, 256] f32 -> out [16, 85] f32.
//
// Stage 1: per-(b, c, by) wave computes the 8 w-bin sums of its 32x256 row
//          block, using V_WMMA_F32_16X16X4_F32 with B=ones as the accumulator
//          (row m of A only ever receives data of w-bin m%8).
// Stage 2: combine 64 channel partials -> s8[8][8] per batch, emit 85 means.

typedef __attribute__((ext_vector_type(2))) float v2f;
typedef __attribute__((ext_vector_type(8))) float v8f;

__global__ __launch_bounds__(32)
void spp_partial_kernel(const float* __restrict__ x, float* __restrict__ partial) {
    const int wg   = blockIdx.x;            // (b*8 + by)*64 + c   -> 0..8191
    const int c    = wg & 63;
    const int by   = (wg >> 6) & 7;
    const int b    = wg >> 9;
    const int lane = threadIdx.x;           // 0..31 (wave32)

    // A-matrix layout (16x4 f32): row M = lane%16, K = 2*(lane/16) + vgpr.
    // All 4 floats of this lane's b128 load belong to row (lane%16).
    // Choose slots so row m only ever sees w-bin m%8.
    const int r    = lane & 15;
    const int half = lane >> 4;
    const int bin  = r & 7;                 // w-bin fed by this lane
    const int u    = (r >> 3) + 2 * half;   // 0..3: which slot pair inside the bin

    const float* row0 = x + (((size_t)(b * 64 + c)) * 256 + (size_t)by * 32) * 256;
    const int off1 = (8 * bin + u) * 4;     // float offset of 1st b128 in a W-row
    const int off2 = off1 + 16;             // +4 slots (= +64 B per bin half)

    v8f acc0 = {};
    v8f acc1 = {};
    const v2f ones = {1.0f, 1.0f};

    #pragma unroll 4
    for (int h = 0; h < 32; ++h) {
        const float* rp = row0 + (size_t)h * 256;
        const float4 a1 = *(const float4*)(rp + off1);
        const float4 a2 = *(const float4*)(rp + off2);
        v2f m0 = {a1.x, a1.y};
        v2f m1 = {a1.z, a1.w};
        v2f m2 = {a2.x, a2.y};
        v2f m3 = {a2.z, a2.w};
        // D[m][n] += sum_k A[m][k]  (B = ones 4x16); two chains to hide XDL latency
        acc0 = __builtin_amdgcn_wmma_f32_16x16x4_f32(false, m0, false, ones, (short)0, acc0, false, false);
        acc1 = __builtin_amdgcn_wmma_f32_16x16x4_f32(false, m1, false, ones, (short)0, acc1, false, false);
        acc0 = __builtin_amdgcn_wmma_f32_16x16x4_f32(false, m2, false, ones, (short)0, acc0, false, false);
        acc1 = __builtin_amdgcn_wmma_f32_16x16x4_f32(false, m3, false, ones, (short)0, acc1, false, false);
    }

    // D columns are identical (B=ones). C/D layout: VGPR j = row j (lanes 0-15)
    // and row j+8 (lanes 16-31); column n = lane%16.
    // Lane L holds D[(L>>4)*8 + j][L&15] in acc[j]. Select j = lane&7, so lane
    // bb (<8) holds D[bb][bb] and lane bb+16 holds D[bb+8][bb]; one xor-16
    // shuffle then gives bin_bb = D[bb][bb] + D[bb+8][bb] on lanes 0..7, which
    // we store with a single coalesced 8-lane write.
    float s[8];
    #pragma unroll
    for (int bb = 0; bb < 8; ++bb) s[bb] = acc0[bb] + acc1[bb];

    const int k = lane & 7;
    float sel = s[0];
    #pragma unroll
    for (int bb = 1; bb < 8; ++bb) sel = (k == bb) ? s[bb] : sel;

    sel += __shfl_xor(sel, 16, 32);
    if (lane < 8) partial[(size_t)wg * 8 + lane] = sel;
}

__global__ __launch_bounds__(256)
void spp_finalize_kernel(const float* __restrict__ partial, float* __restrict__ out) {
    const int b = blockIdx.x;       // 0..15
    const int t = threadIdx.x;      // 0..255

    __shared__ float red[256];
    __shared__ float s8[64];        // [by][bx] sums over C and 32x32 window

    // Sum the 64 channel partials for each (by, bx); 4-way split over threads.
    const int by = (t & 63) >> 3;
    const int bx = t & 7;
    const int q  = t >> 6;
    float s = 0.0f;
    for (int i = 0; i < 16; ++i) {
        const int c = q * 16 + i;
        s += partial[((size_t)((b * 8 + by) * 64 + c)) * 8 + bx];
    }
    red[t] = s;
    __syncthreads();
    if (t < 64) s8[t] = red[t] + red[t + 64] + red[t + 128] + red[t + 192];
    __syncthreads();

    // Output layout per batch: [lvl1 (1) | lvl2 (4) | lvl4 (16) | lvl8 (64)]
    if (t < 85) {
        float v = 0.0f;
        float scale;
        if (t == 0) {
            for (int i = 0; i < 64; ++i) v += s8[i];
            scale = 1.0f / (65536.0f * 64.0f);
        } else if (t < 5) {                       // level 2: 4x4 blocks of s8
            const int idx = t - 1, i0 = (idx >> 1) * 4, j0 = (idx & 1) * 4;
            for (int di = 0; di < 4; ++di)
                for (int dj = 0; dj < 4; ++dj) v += s8[(i0 + di) * 8 + (j0 + dj)];
            scale = 1.0f / (65536.0f * 16.0f);
        } else if (t < 21) {                      // level 4: 2x2 blocks of s8
            const int idx = t - 5, i0 = (idx >> 2) * 2, j0 = (idx & 3) * 2;
            for (int di = 0; di < 2; ++di)
                for (int dj = 0; dj < 2; ++dj) v += s8[(i0 + di) * 8 + (j0 + dj)];
            scale = 1.0f / (65536.0f * 4.0f);
        } else {                                  // level 8: s8 itself
            v = s8[t - 21];
            scale = 1.0f / 65536.0f;
        }
        out[b * 85 + t] = v * scale;
    }
}

extern "C" void kernel_launch(void* const* d_in, const int* in_sizes, int n_in,
                              void* d_out, int out_size, void* d_ws, size_t ws_size,
                              hipStream_t stream) {
    const float* x  = (const float*)d_in[0];   // [16, 64, 256, 256] f32
    float* partial  = (float*)d_ws;            // 8192 * 8 floats = 256 KB scratch
    float* out      = (float*)d_out;           // [16, 85] f32

    spp_partial_kernel<<<dim3(16 * 8 * 64), dim3(32), 0, stream>>>(x, partial);
    spp_finalize_kernel<<<dim3(16), dim3(256), 0, stream>>>(partial, out);
}